// Net_40123584479271
// MI455X (gfx1250) — compile-verified
//
#include <hip/hip_runtime.h>

typedef float v2f __attribute__((ext_vector_type(2)));
typedef float v8f __attribute__((ext_vector_type(8)));

#define IN_DIM 256
#define HID 16
#define OUTD 10

// -------- init: agg1 = 0, out[i*10+o] = b2[o] (bias pre-seeded so scatter2 lands on top)
__global__ void gcn_init_kernel(float* __restrict__ agg, float* __restrict__ out,
                                const float* __restrict__ b2, int n_nodes) {
    int idx = blockIdx.x * blockDim.x + threadIdx.x;
    if (idx < n_nodes * HID)  agg[idx] = 0.0f;
    if (idx < n_nodes * OUTD) out[idx] = b2[idx % OUTD];
}

// -------- GEMM1: m1 = x @ W1 via V_WMMA_F32_16X16X4_F32, one wave per 16-row tile
__global__ void gcn_gemm1_wmma(const float* __restrict__ x, const float* __restrict__ W1,
                               float* __restrict__ m1, int n_tiles) {
    int wave = (int)((blockIdx.x * blockDim.x + threadIdx.x) >> 5);
    int lane = threadIdx.x & 31;
    if (wave >= n_tiles) return;          // wave-uniform: EXEC stays all-1s below
    int l15 = lane & 15;                  // A: row (M); B: col (N)
    int hi  = lane >> 4;                  // lane-half selects K sub-pair
    const float* xrow = x + ((size_t)(wave * 16 + l15)) * IN_DIM;

    v8f c = {};
    #pragma unroll 4
    for (int k0 = 0; k0 < IN_DIM; k0 += 4) {
        int ka = k0 + hi * 2;
        v2f a; a.x = xrow[ka];                 a.y = xrow[ka + 1];
        v2f b; b.x = W1[(ka + 0) * HID + l15]; b.y = W1[(ka + 1) * HID + l15];
        c = __builtin_amdgcn_wmma_f32_16x16x4_f32(false, a, false, b, (short)0, c, false, false);
    }
    float* mt = m1 + (size_t)(wave * 16) * HID;
    #pragma unroll
    for (int v = 0; v < 8; ++v)
        mt[(v + hi * 8) * HID + l15] = c[v];   // D: M = v + hi*8, N = l15
}

// -------- scatter1: agg[dst] += m1[src], 16 floats per edge (L2-resident atomics)
__global__ void gcn_scatter1(const int* __restrict__ src, const int* __restrict__ dst,
                             const float* __restrict__ m1, float* __restrict__ agg,
                             long long n_edges) {
    long long e = (long long)blockIdx.x * blockDim.x + threadIdx.x;
    if (e >= n_edges) return;
    int s = src[e], d = dst[e];
    const float4* ms = (const float4*)(m1 + (size_t)s * HID);
    float* ad = agg + (size_t)d * HID;
    #pragma unroll
    for (int q = 0; q < 4; ++q) {
        float4 v = ms[q];
        atomicAdd(ad + q * 4 + 0, v.x);
        atomicAdd(ad + q * 4 + 1, v.y);
        atomicAdd(ad + q * 4 + 2, v.z);
        atomicAdd(ad + q * 4 + 3, v.w);
    }
}

// -------- GEMM2: m2 = relu(agg + b1) @ W2 (W2 zero-padded 16x10 -> 16x16), K=16 -> 4 WMMAs
__global__ void gcn_gemm2_wmma(const float* __restrict__ agg, const float* __restrict__ b1,
                               const float* __restrict__ W2, float* __restrict__ m2,
                               int n_tiles) {
    int wave = (int)((blockIdx.x * blockDim.x + threadIdx.x) >> 5);
    int lane = threadIdx.x & 31;
    if (wave >= n_tiles) return;
    int l15 = lane & 15;
    int hi  = lane >> 4;
    const float* arow = agg + (size_t)(wave * 16 + l15) * HID;

    v8f c = {};
    #pragma unroll
    for (int k0 = 0; k0 < HID; k0 += 4) {
        int ka = k0 + hi * 2;
        v2f a;
        a.x = fmaxf(arow[ka]     + b1[ka],     0.0f);   // fused relu(+b1) on A-load
        a.y = fmaxf(arow[ka + 1] + b1[ka + 1], 0.0f);
        v2f b;
        b.x = (l15 < OUTD) ? W2[(ka + 0) * OUTD + l15] : 0.0f;
        b.y = (l15 < OUTD) ? W2[(ka + 1) * OUTD + l15] : 0.0f;
        c = __builtin_amdgcn_wmma_f32_16x16x4_f32(false, a, false, b, (short)0, c, false, false);
    }
    float* mt = m2 + (size_t)(wave * 16) * HID;   // padded N x 16; cols 10..15 are exact zeros
    #pragma unroll
    for (int v = 0; v < 8; ++v)
        mt[(v + hi * 8) * HID + l15] = c[v];
}

// -------- scatter2: out[dst] += m2[src], 10 floats per edge (out pre-seeded with b2)
__global__ void gcn_scatter2(const int* __restrict__ src, const int* __restrict__ dst,
                             const float* __restrict__ m2, float* __restrict__ out,
                             long long n_edges) {
    long long e = (long long)blockIdx.x * blockDim.x + threadIdx.x;
    if (e >= n_edges) return;
    int s = src[e], d = dst[e];
    const float* ms = m2 + (size_t)s * HID;   // padded row
    float* od = out + (size_t)d * OUTD;
    #pragma unroll
    for (int q = 0; q < OUTD; ++q)
        atomicAdd(od + q, ms[q]);
}

extern "C" void kernel_launch(void* const* d_in, const int* in_sizes, int n_in,
                              void* d_out, int out_size, void* d_ws, size_t ws_size,
                              hipStream_t stream) {
    const float* x  = (const float*)d_in[0];
    const int*   ei = (const int*)  d_in[1];
    const float* W1 = (const float*)d_in[2];
    const float* b1 = (const float*)d_in[3];
    const float* W2 = (const float*)d_in[4];
    const float* b2 = (const float*)d_in[5];
    float* out = (float*)d_out;

    int n_nodes       = in_sizes[0] / IN_DIM;           // 100000
    long long n_edges = (long long)in_sizes[1] / 2;     // 3200000
    const int* src = ei;                                 // edge_index row 0
    const int* dst = ei + n_edges;                       // edge_index row 1

    // workspace layout (all fit in L2): m1 | agg | m2  (each n_nodes*16 floats)
    float* m1  = (float*)d_ws;
    float* agg = m1  + (size_t)n_nodes * HID;
    float* m2  = agg + (size_t)n_nodes * HID;

    int n_tiles = (n_nodes + 15) / 16;                   // 6250 exact tiles
    int init_blocks = (n_nodes * HID + 255) / 256;
    int gemm_blocks = (n_tiles + 7) / 8;                 // 8 waves (tiles) per 256-thread block
    int edge_blocks = (int)((n_edges + 255) / 256);

    gcn_init_kernel<<<init_blocks, 256, 0, stream>>>(agg, out, b2, n_nodes);
    gcn_gemm1_wmma <<<gemm_blocks, 256, 0, stream>>>(x, W1, m1, n_tiles);
    gcn_scatter1   <<<edge_blocks, 256, 0, stream>>>(src, dst, m1, agg, n_edges);
    gcn_gemm2_wmma <<<gemm_blocks, 256, 0, stream>>>(agg, b1, W2, m2, n_tiles);
    gcn_scatter2   <<<edge_blocks, 256, 0, stream>>>(src, dst, m2, out, n_edges);
}